// BiGCNModel_7069516169810
// MI455X (gfx1250) — compile-verified
//
#include <hip/hip_runtime.h>
#include <cmath>

// ---------------------------------------------------------------------------
// GCNII on MI455X (gfx1250, wave32). fp32 WMMA 16x16x4 for all GEMMs
// (workload is L2/atomic bound, so full-precision matrix math is free).
// ---------------------------------------------------------------------------

typedef __attribute__((ext_vector_type(2))) float v2f;
typedef __attribute__((ext_vector_type(8))) float v8f;

#define CCH 128   // channels
#define TILE 16

// ---------------------------------------------------------------- utilities
__global__ void zero_kernel(float* __restrict__ p, int n) {
  int i = blockIdx.x * blockDim.x + threadIdx.x;
  if (i < n) p[i] = 0.0f;
}

// ------------------------------------------------ encoder: Y = relu(X@W + b)
// X[N,128], W[128,128], one block = 8 waves = 16 rows x 128 cols of output.
__global__ void gemm_bias_relu(const float* __restrict__ X,
                               const float* __restrict__ W,
                               const float* __restrict__ bias,
                               float* __restrict__ Y) {
  const int wave = threadIdx.x >> 5;
  const int lane = threadIdx.x & 31;
  const int rowBase = blockIdx.x * TILE;
  const int colBase = wave * TILE;
  const int m  = lane & 15;           // A: row within tile / B,D: col within tile
  const int kh = (lane >> 4) << 1;    // K sub-offset for upper lane half
  const float* __restrict__ xrow = X + (size_t)(rowBase + m) * CCH;
  v8f acc = {};
#pragma unroll
  for (int k = 0; k < CCH; k += 4) {
    v2f a, b;
    a.x = xrow[k + kh];
    a.y = xrow[k + kh + 1];
    b.x = W[(size_t)(k + kh)     * CCH + colBase + m];
    b.y = W[(size_t)(k + kh + 1) * CCH + colBase + m];
    acc = __builtin_amdgcn_wmma_f32_16x16x4_f32(false, a, false, b,
                                                (short)0, acc, false, false);
  }
  const int col = colBase + m;
  const float bb = bias[col];
  const int row0 = rowBase + ((lane >> 4) << 3);
#pragma unroll
  for (int v = 0; v < 8; ++v) {
    float val = acc[v] + bb;
    Y[(size_t)(row0 + v) * CCH + col] = fmaxf(val, 0.0f);
  }
}

// -------------------------------- layer: H = (1-beta)*S + beta*(S @ W) -----
__global__ void gemm_layer(const float* __restrict__ S,
                           const float* __restrict__ W,
                           float* __restrict__ H,
                           float beta) {
  const int wave = threadIdx.x >> 5;
  const int lane = threadIdx.x & 31;
  const int rowBase = blockIdx.x * TILE;
  const int colBase = wave * TILE;
  const int m  = lane & 15;
  const int kh = (lane >> 4) << 1;
  const float* __restrict__ srow = S + (size_t)(rowBase + m) * CCH;
  v8f acc = {};
#pragma unroll
  for (int k = 0; k < CCH; k += 4) {
    v2f a, b;
    a.x = srow[k + kh];
    a.y = srow[k + kh + 1];
    b.x = W[(size_t)(k + kh)     * CCH + colBase + m];
    b.y = W[(size_t)(k + kh + 1) * CCH + colBase + m];
    acc = __builtin_amdgcn_wmma_f32_16x16x4_f32(false, a, false, b,
                                                (short)0, acc, false, false);
  }
  const int col = colBase + m;
  const int row0 = rowBase + ((lane >> 4) << 3);
  const float omb = 1.0f - beta;
#pragma unroll
  for (int v = 0; v < 8; ++v) {
    size_t idx = (size_t)(row0 + v) * CCH + col;
    H[idx] = omb * S[idx] + beta * acc[v];
  }
}

// ---------------------------- head GEMM: Z[N,16] = H[N,128] @ W1[128,16] + b1
// one wave per 16-row tile (single 16-wide column tile); wave-uniform guard.
__global__ void gemm_head(const float* __restrict__ H,
                          const float* __restrict__ W1,
                          const float* __restrict__ b1,
                          float* __restrict__ Z, int nTiles) {
  const int wave = threadIdx.x >> 5;
  const int tile = blockIdx.x * 8 + wave;
  if (tile >= nTiles) return;   // wave-uniform: EXEC stays all-1s inside
  const int lane = threadIdx.x & 31;
  const int rowBase = tile * TILE;
  const int m  = lane & 15;
  const int kh = (lane >> 4) << 1;
  const float* __restrict__ xrow = H + (size_t)(rowBase + m) * CCH;
  v8f acc = {};
#pragma unroll
  for (int k = 0; k < CCH; k += 4) {
    v2f a, b;
    a.x = xrow[k + kh];
    a.y = xrow[k + kh + 1];
    b.x = W1[(size_t)(k + kh)     * 16 + m];
    b.y = W1[(size_t)(k + kh + 1) * 16 + m];
    acc = __builtin_amdgcn_wmma_f32_16x16x4_f32(false, a, false, b,
                                                (short)0, acc, false, false);
  }
  const float bb = b1[m];
  const int row0 = rowBase + ((lane >> 4) << 3);
#pragma unroll
  for (int v = 0; v < 8; ++v)
    Z[(size_t)(row0 + v) * 16 + m] = acc[v] + bb;
}

// -------------------------------- edge scatter-add: agg[dst] += h[src] -----
// one wave per edge; lane handles 4 channels (float4 gather, 4 f32 atomics).
// h/agg are 51 MB each -> resident in the 192 MB L2; atomics hit L2.
__global__ void aggregate_kernel(const float* __restrict__ h,
                                 const int* __restrict__ src,
                                 const int* __restrict__ dst,
                                 float* __restrict__ agg, int E) {
  const int e = blockIdx.x * 8 + (threadIdx.x >> 5);
  if (e >= E) return;
  const int lane = threadIdx.x & 31;
  const int s = src[e];
  const int d = dst[e];
  const float4 v = ((const float4*)(h + (size_t)s * CCH))[lane];
  float* ap = agg + (size_t)d * CCH + lane * 4;
  atomicAdd(ap + 0, v.x);
  atomicAdd(ap + 1, v.y);
  atomicAdd(ap + 2, v.z);
  atomicAdd(ap + 3, v.w);
}

// --------------------------- support = (1-alpha)*agg + alpha*x0 (in place) -
__global__ void support_kernel(float* __restrict__ agg,
                               const float* __restrict__ x0, int n) {
  int i = blockIdx.x * blockDim.x + threadIdx.x;
  if (i < n) agg[i] = 0.9f * agg[i] + 0.1f * x0[i];
}

// ------------------- per-channel sum / sumsq (stats[0:C]=sum, [C:2C]=sumsq) -
// requires (gridDim*blockDim) % Ccols == 0.
__global__ void colstats_kernel(const float* __restrict__ X,
                                float* __restrict__ stats, int N, int Ccols) {
  const int tid = blockIdx.x * blockDim.x + threadIdx.x;
  const int c = tid % Ccols;
  const int stride = (gridDim.x * blockDim.x) / Ccols;
  float s = 0.0f, sq = 0.0f;
  for (int r = tid / Ccols; r < N; r += stride) {
    float v = X[(size_t)r * Ccols + c];
    s += v;
    sq += v * v;
  }
  atomicAdd(&stats[c], s);
  atomicAdd(&stats[Ccols + c], sq);
}

// ------------------------------ h = relu(batchnorm(h)) using stats ---------
__global__ void bn_relu_kernel(float* __restrict__ h,
                               const float* __restrict__ stats,
                               const float* __restrict__ g,
                               const float* __restrict__ b,
                               int N, int n) {
  int i = blockIdx.x * blockDim.x + threadIdx.x;
  if (i >= n) return;
  const int c = i & (CCH - 1);
  const float invN = 1.0f / (float)N;
  const float mean = stats[c] * invN;
  const float var  = stats[CCH + c] * invN - mean * mean;
  const float sc   = rsqrtf(var + 1e-5f) * g[c];
  h[i] = fmaxf((h[i] - mean) * sc + b[c], 0.0f);
}

// ----------------- head: out = batchnorm(Z) @ w2 + b2  (16 -> 1) -----------
__global__ void head_out_kernel(const float* __restrict__ Z,
                                const float* __restrict__ stats,
                                const float* __restrict__ g,
                                const float* __restrict__ b,
                                const float* __restrict__ w2,
                                const float* __restrict__ b2,
                                float* __restrict__ out, int N) {
  int i = blockIdx.x * blockDim.x + threadIdx.x;
  if (i >= N) return;
  const float invN = 1.0f / (float)N;
  float acc = 0.0f;
#pragma unroll
  for (int c = 0; c < 16; ++c) {
    float mean = stats[c] * invN;
    float var  = stats[16 + c] * invN - mean * mean;
    float zn = (Z[(size_t)i * 16 + c] - mean) * rsqrtf(var + 1e-5f) * g[c] + b[c];
    acc += zn * w2[c];
  }
  out[i] = acc + b2[0];
}

// ---------------------------------------------------------------------------
extern "C" void kernel_launch(void* const* d_in, const int* in_sizes, int n_in,
                              void* d_out, int out_size, void* d_ws, size_t ws_size,
                              hipStream_t stream) {
  const float* x     = (const float*)d_in[0];
  const int*   ei    = (const int*)  d_in[1];
  const float* lin_w = (const float*)d_in[2];
  const float* lin_b = (const float*)d_in[3];
  const float* convw = (const float*)d_in[4];
  const float* bn1g  = (const float*)d_in[5];
  const float* bn1b  = (const float*)d_in[6];
  const float* l1w   = (const float*)d_in[7];
  const float* l1b   = (const float*)d_in[8];
  const float* bn2g  = (const float*)d_in[9];
  const float* bn2b  = (const float*)d_in[10];
  const float* l2w   = (const float*)d_in[11];
  const float* l2b   = (const float*)d_in[12];
  float* out = (float*)d_out;

  const int N = in_sizes[0] / CCH;  // 100000 (multiple of 16)
  const int E = in_sizes[1] / 2;    // 1600000
  const int NC = N * CCH;

  // workspace layout (floats): x0 | h | agg (head Z reuses agg) | stats(512)
  float* x0    = (float*)d_ws;
  float* h     = x0  + (size_t)NC;
  float* agg   = h   + (size_t)NC;
  float* stats = agg + (size_t)NC;

  const int rowTiles = N / TILE;                // 6250 exact tiles
  const int ewBlocks = (E + 7) / 8;             // 8 edge-waves per block
  const int nBlocks  = (NC + 255) / 256;

  // encoder: x0 = relu(x @ lin_w + lin_b)
  gemm_bias_relu<<<rowTiles, 256, 0, stream>>>(x, lin_w, lin_b, x0);

  const float* hin = x0;
  for (int l = 0; l < 4; ++l) {
    const float beta = logf(0.5f / (float)(l + 1) + 1.0f);
    zero_kernel<<<nBlocks, 256, 0, stream>>>(agg, NC);
    aggregate_kernel<<<ewBlocks, 256, 0, stream>>>(hin, ei, ei + E, agg, E);
    support_kernel<<<nBlocks, 256, 0, stream>>>(agg, x0, NC);
    gemm_layer<<<rowTiles, 256, 0, stream>>>(agg, convw + (size_t)l * CCH * CCH, h, beta);
    zero_kernel<<<1, 256, 0, stream>>>(stats, 2 * CCH);
    colstats_kernel<<<256, 256, 0, stream>>>(h, stats, N, CCH);   // 65536 % 128 == 0
    bn_relu_kernel<<<nBlocks, 256, 0, stream>>>(h, stats, bn1g, bn1b, N, NC);
    hin = h;
  }

  // head
  float* z = agg;            // reuse
  float* stats2 = stats + 2 * CCH;
  gemm_head<<<(rowTiles + 7) / 8, 256, 0, stream>>>(h, l1w, l1b, z, rowTiles);
  zero_kernel<<<1, 64, 0, stream>>>(stats2, 32);
  colstats_kernel<<<64, 256, 0, stream>>>(z, stats2, N, 16);      // 16384 % 16 == 0
  head_out_kernel<<<(N + 255) / 256, 256, 0, stream>>>(z, stats2, bn2g, bn2b,
                                                       l2w, l2b, out, N);
}